// LogEig_43215960933253
// MI455X (gfx1250) — compile-verified
//
#include <hip/hip_runtime.h>
#include <math.h>

// Batched 64x64 SPD matrix logarithm:  out = logm(X) = V log(Lam) V^T
// via inverse scaling-and-squaring + coupled Newton-Schulz square roots.
// All 64x64 fp32 GEMMs run on V_WMMA_F32_16X16X4_F32 with LDS-resident data.
// All intermediate matrices are symmetric (polynomials in SPD X), so the
// B operand is read row-major (B[n][k]) -> contiguous b64 LDS loads like A.

typedef __attribute__((ext_vector_type(2))) float v2f;
typedef __attribute__((ext_vector_type(8))) float v8f;
typedef __attribute__((ext_vector_type(4))) unsigned int u32x4;
typedef __attribute__((ext_vector_type(4))) int i32x4;
typedef __attribute__((ext_vector_type(8))) int i32x8;

#define MATN 64
#define LDA  68          // padded LDS stride (floats): 4-bank skew, rows 16B-aligned
#define NTHREADS 256     // 8 waves (wave32)

#if defined(__has_builtin)
#if __has_builtin(__builtin_amdgcn_tensor_load_to_lds) && \
    __has_builtin(__builtin_amdgcn_s_wait_tensorcnt)
#define HAVE_TDM 1
#endif
#endif
#ifndef HAVE_TDM
#define HAVE_TDM 0
#endif

// C = alpha*(A x B) + beta_diag*I, A,B,C 64x64 fp32 in LDS (stride LDA),
// B assumed symmetric (B[k][n] == B[n][k]) so both operands load contiguously.
__device__ __forceinline__ void gemm64_sym(const float* A, const float* B, float* C,
                                           float alpha, float beta_diag,
                                           int lane, int wave)
{
    const int row16  = lane & 15;        // M within tile (A/C) or N within tile (B/C)
    const int koff   = (lane >> 4) * 2;  // K sub-offset per ISA A/B layout (0 or 2)
    const int hshift = (lane >> 4) * 8;  // C/D row offset for upper half-wave

    const int t0  = wave * 2;            // two tiles per wave, same mi -> A reuse
    const int mi  = t0 >> 2;
    const int ni0 = t0 & 3;
    const int ni1 = ni0 + 1;

    const int arow  = mi  * 16 + row16;  // A panel row
    const int brow0 = ni0 * 16 + row16;  // B read as B[n][k] (symmetry)
    const int brow1 = ni1 * 16 + row16;

    v8f acc0 = {};
    v8f acc1 = {};

#pragma unroll
    for (int kt = 0; kt < 16; ++kt) {
        const int k0 = kt * 4 + koff;
        const v2f a  = *(const v2f*)&A[arow  * LDA + k0];
        const v2f b0 = *(const v2f*)&B[brow0 * LDA + k0];
        const v2f b1 = *(const v2f*)&B[brow1 * LDA + k0];
        acc0 = __builtin_amdgcn_wmma_f32_16x16x4_f32(false, a, false, b0,
                                                     (short)0, acc0, false, false);
        acc1 = __builtin_amdgcn_wmma_f32_16x16x4_f32(false, a, false, b1,
                                                     (short)0, acc1, false, false);
    }

#pragma unroll
    for (int r = 0; r < 8; ++r) {
        const int row  = mi * 16 + hshift + r;
        const int col0 = ni0 * 16 + row16;
        const int col1 = ni1 * 16 + row16;
        C[row * LDA + col0] = alpha * acc0[r] + ((row == col0) ? beta_diag : 0.0f);
        C[row * LDA + col1] = alpha * acc1[r] + ((row == col1) ? beta_diag : 0.0f);
    }
}

__device__ __forceinline__ void set_scaled_identity(float* M, float d, int tid)
{
    for (int i = tid; i < MATN * MATN; i += NTHREADS) {
        const int r = i >> 6, c = i & 63;
        M[r * LDA + c] = (r == c) ? d : 0.0f;
    }
}

__global__ void __launch_bounds__(NTHREADS)
logm64_wmma_kernel(const float* __restrict__ X, float* __restrict__ OUT)
{
    __shared__ float buf[4][MATN * LDA];
    __shared__ float red;

    const int tid  = threadIdx.x;
    const int lane = tid & 31;
    const int wave = tid >> 5;

    const float* Xb = X   + (size_t)blockIdx.x * MATN * MATN;
    float*       Ob = OUT + (size_t)blockIdx.x * MATN * MATN;

    if (tid == 0) red = 0.0f;

    // ---- X -> LDS (padded layout), via Tensor Data Mover when available ----
#if HAVE_TDM
    if (wave == 0) {
        const unsigned long long ga = (unsigned long long)(const void*)Xb;
        const unsigned int lds_base = (unsigned int)(unsigned long long)(const void*)&buf[0][0];
        u32x4 g0;
        g0[0] = 1u;                                    // count=1 (valid), user mode
        g0[1] = lds_base;                              // lds_addr
        g0[2] = (unsigned int)ga;                      // global_addr[31:0]
        g0[3] = (unsigned int)(ga >> 32) | (2u << 30); // global_addr[56:32] | type=2
        i32x8 g1;
        g1[0] = (2 << 16)        // data_size = 4 bytes
              | (1 << 20)        // pad_enable
              | (5 << 22)        // pad_interval: every 64 DWORDs
              | (3 << 25);       // pad_amount: 4 DWORDs  -> LDS stride 68 floats
        g1[1] = MATN << 16;      // tensor_dim0 (bits[79:48])
        g1[2] = MATN << 16;      // tensor_dim1 (bits[111:80])
        g1[3] = MATN << 16;      // tile_dim0   (bits[127:112])
        g1[4] = MATN;            // tile_dim1   (bits[143:128])
        g1[5] = MATN;            // tensor_dim0_stride (bits[207:160])
        g1[6] = 0;
        g1[7] = 0;
        const i32x4 z4 = {0, 0, 0, 0};
#if __has_include(<hip/amd_detail/amd_gfx1250_TDM.h>)
        const i32x8 z8 = {0, 0, 0, 0, 0, 0, 0, 0};
        __builtin_amdgcn_tensor_load_to_lds(g0, g1, z4, z4, z8, 0);
#else
        __builtin_amdgcn_tensor_load_to_lds(g0, g1, z4, z4, 0);
#endif
        __builtin_amdgcn_s_wait_tensorcnt(0);
    }
#else
    for (int i = tid * 4; i < MATN * MATN; i += NTHREADS * 4) {
        const int r = i >> 6, c = i & 63;
        *(float4*)&buf[0][r * LDA + c] = *(const float4*)(Xb + i);
    }
#endif
    __syncthreads();

    // ---- Frobenius norm^2 reduction (>= lambda_max^2 for SPD) --------------
    float part = 0.0f;
    for (int i = tid * 4; i < MATN * MATN; i += NTHREADS * 4) {
        const int r = i >> 6, c = i & 63;
        const float4 v = *(const float4*)&buf[0][r * LDA + c];
        part += v.x * v.x + v.y * v.y + v.z * v.z + v.w * v.w;
    }
    atomicAdd(&red, part);              // ds_add_f32
    __syncthreads();

    const float s     = sqrtf(red);
    const float inv_s = 1.0f / s;

    // Y = X / s : spectrum in (0,1], Newton-Schulz domain
    for (int i = tid; i < MATN * MATN; i += NTHREADS) {
        const int r = i >> 6, c = i & 63;
        buf[0][r * LDA + c] *= inv_s;
    }
    __syncthreads();

    int yi = 0, zi = 1, wi = 2, ti = 3;

    // ---- 5 sqrt stages, coupled Newton-Schulz ------------------------------
    const int iters[5] = {14, 11, 8, 6, 5};
    for (int stage = 0; stage < 5; ++stage) {
        set_scaled_identity(buf[zi], 1.0f, tid);          // Z = I
        __syncthreads();
        const int nit = iters[stage];
        for (int it = 0; it < nit; ++it) {
            gemm64_sym(buf[zi], buf[yi], buf[wi], -0.5f, 1.5f, lane, wave); // W = 1.5I - 0.5 ZY
            __syncthreads();
            gemm64_sym(buf[yi], buf[wi], buf[ti], 1.0f, 0.0f, lane, wave);  // T = Y W   (new Y)
            __syncthreads();
            gemm64_sym(buf[wi], buf[zi], buf[yi], 1.0f, 0.0f, lane, wave);  // oldY = W Z (new Z)
            __syncthreads();
            const int ny = ti, nz = yi, nt = zi;   // rotate buffers (W reused)
            yi = ny; zi = nz; ti = nt;
        }
    }

    // ---- log series: Y = A^(1/32) = I + E, ||E|| small ---------------------
    for (int i = tid; i < MATN * MATN; i += NTHREADS) {
        const int r = i >> 6, c = i & 63;
        buf[zi][r * LDA + c] = buf[yi][r * LDA + c] - ((r == c) ? 1.0f : 0.0f);
    }
    __syncthreads();

    // Horner: log(I+E) = E*(c1 I + E*(c2 I + ... + E*(c8 I)...)), cm = (-1)^(m+1)/m
    set_scaled_identity(buf[wi], -1.0f / 8.0f, tid);      // S = c8 I
    __syncthreads();
    for (int m = 7; m >= 1; --m) {
        const float cm = ((m & 1) ? 1.0f : -1.0f) / (float)m;
        gemm64_sym(buf[zi], buf[wi], buf[ti], 1.0f, cm, lane, wave);  // T = E*S + cm I
        __syncthreads();
        const int t = wi; wi = ti; ti = t;                            // S <- T
    }

    // OUT = 2^5 * (E*S) + ln(s) * I  (Y buffer dead -> output buffer)
    gemm64_sym(buf[zi], buf[wi], buf[yi], 32.0f, logf(s), lane, wave);
    __syncthreads();

    // ---- store -------------------------------------------------------------
    for (int i = tid * 4; i < MATN * MATN; i += NTHREADS * 4) {
        const int r = i >> 6, c = i & 63;
        *(float4*)(Ob + i) = *(const float4*)&buf[yi][r * LDA + c];
    }
}

extern "C" void kernel_launch(void* const* d_in, const int* in_sizes, int n_in,
                              void* d_out, int out_size, void* d_ws, size_t ws_size,
                              hipStream_t stream)
{
    (void)n_in; (void)d_ws; (void)ws_size; (void)out_size;
    const float* X = (const float*)d_in[0];
    float* OUT = (float*)d_out;
    const int B = in_sizes[0] / (MATN * MATN);   // 8192
    logm64_wmma_kernel<<<B, NTHREADS, 0, stream>>>(X, OUT);
}